// ViewBuffer_76836964926276
// MI455X (gfx1250) — compile-verified
//
#include <hip/hip_runtime.h>
#include <stdint.h>

// Problem constants from the reference: (NUM_NODES, K, DIM, B) = (200000, 8, 128, 65536)
constexpr int K   = 8;
constexpr int DIM = 128;

typedef float v4f __attribute__((ext_vector_type(4)));

// One block (256 threads = 8 wave32) per batch element b.
//   thread (k = tid>>5, j = tid&31) copies float4 #j of row-slot (k+p)%8.
// Wave-coalesced: 32 lanes x 16B = one contiguous 512B row segment per wave,
// for both the gathered load and the streaming store.
__global__ __launch_bounds__(256) void viewbuffer_gather_kernel(
    const float* __restrict__ emb,    // (NUM_NODES, K, DIM) f32
    const float* __restrict__ timeb,  // (NUM_NODES, K)      f32
    const int*   __restrict__ idx,    // (B,)                i32
    const int*   __restrict__ ptrb,   // (NUM_NODES,)        i32
    float* __restrict__ out_views,    // (B, K, DIM)
    float* __restrict__ out_times,    // (B, K)
    float* __restrict__ out_dts)      // (B, K)
{
    __shared__ float s_times[K];

    const int b   = blockIdx.x;
    const int tid = threadIdx.x;

    // Uniform per-block scalars -> SMEM loads.
    const int node = idx[b];
    const int p    = ptrb[node];          // non-negative, so & (K-1) == % K

    const int k    = tid >> 5;            // 0..7  (wave id within block)
    const int j    = tid & 31;            // 0..31 (lane id)
    const int slot = (k + p) & (K - 1);   // circular roll: oldest-first

    // ---- bulk views copy: gather 512B rows, stream out non-temporally ----
    const v4f* src = reinterpret_cast<const v4f*>(
        emb + ((size_t)node * K + slot) * DIM) + j;
    v4f v = *src;                          // global_load_b128, RT (L2 reuse on dup idx)

    v4f* dst = reinterpret_cast<v4f*>(
        out_views + ((size_t)b * K + k) * DIM) + j;
    __builtin_nontemporal_store(v, dst);   // global_store_b128 TH=NT (write-once stream)

    // ---- times / dts: 8 values via CDNA5 async load-to-LDS (ASYNCcnt) ----
    // Lanes 0..7 of wave 0 each pull one f32 straight into LDS (no VGPR hop),
    // wait the async counter in the same wave, then compute the clamped
    // neighbor difference. Only wave 0 touches LDS -> no workgroup barrier.
    if (tid < K) {
        const int kslot  = (tid + p) & (K - 1);
        uint32_t lds_off = (uint32_t)(uintptr_t)(&s_times[tid]);  // flat->LDS: low 32 bits
        uint64_t gaddr   = (uint64_t)(uintptr_t)(timeb + (size_t)node * K + kslot);

        asm volatile("global_load_async_to_lds_b32 %0, %1, off"
                     :: "v"(lds_off), "v"(gaddr)
                     : "memory");
        asm volatile("s_wait_asynccnt 0" ::: "memory");

        const float t  = s_times[tid];
        const float tp = (tid == 0) ? t : s_times[tid - 1];
        float dt = t - tp;
        dt = dt > 0.0f ? dt : 0.0f;

        out_times[(size_t)b * K + tid] = t;
        out_dts  [(size_t)b * K + tid] = (tid == 0) ? 0.0f : dt;
    }
}

extern "C" void kernel_launch(void* const* d_in, const int* in_sizes, int n_in,
                              void* d_out, int out_size, void* d_ws, size_t ws_size,
                              hipStream_t stream) {
    // setup_inputs() order: emb_buf, time_buf, idx, ptr
    const float* emb   = (const float*)d_in[0];
    const float* timeb = (const float*)d_in[1];
    const int*   idx   = (const int*)d_in[2];
    const int*   ptrb  = (const int*)d_in[3];

    const int B = in_sizes[2];  // 65536

    // Output tuple flattened in return order: views (B,K,DIM), times (B,K), dts (B,K)
    float* out_views = (float*)d_out;
    float* out_times = out_views + (size_t)B * K * DIM;
    float* out_dts   = out_times + (size_t)B * K;

    viewbuffer_gather_kernel<<<B, 256, 0, stream>>>(
        emb, timeb, idx, ptrb, out_views, out_times, out_dts);
}